// EmbeddedDropout_17454747091464
// MI455X (gfx1250) — compile-verified
//
#include <hip/hip_runtime.h>

// EmbeddedDropout: out[t, :] = mask(words[t]) * weight[words[t], :]
//   words      : int32  [B*S]        (B=4, S=4096 -> 16384 tokens)
//   weight     : float32[V, D]       (V=50257, D=1024)
//   row_mask_u : float32[V, 1]       uniform draws; keep if u < 0.9, scale 1/0.9
//   out        : float32[B*S, D]
//
// Pure gather / bandwidth kernel: ~64 MiB read + ~64 MiB write -> ~5.7 us at
// 23.3 TB/s. Strategy: one 4 KB row per block iteration, B128 loads, NT B128
// stores (output is write-once; keep L2 for repeated vocab rows), and a
// global_prefetch of the NEXT token's row to pipeline the gather latency.

typedef float v4f __attribute__((ext_vector_type(4)));

#ifndef EMB_D
#define EMB_D 1024
#endif

__global__ __launch_bounds__(256) void embedded_dropout_gather(
    const int* __restrict__ words,
    const float* __restrict__ weight,
    const float* __restrict__ row_mask_u,
    float* __restrict__ out,
    int n_tokens)
{
    const int tid    = threadIdx.x;   // 0..255 ; 256 lanes * 16 B = 4 KB = one row
    const int stride = gridDim.x;

    const float keep     = 0.9f;      // 1 - DROPOUT
    const float inv_keep = 1.0f / 0.9f;

    for (int t = blockIdx.x; t < n_tokens; t += stride) {
        // Row index for this token (block-uniform -> scalar load expected).
        const int row = words[t];

        // Software pipeline: prefetch next iteration's row into cache while
        // we stream this one. Lowers to global_prefetch_b8 on gfx1250.
        const int tn = t + stride;
        if (tn < n_tokens) {
            const int nrow = words[tn];
            const char* pf = reinterpret_cast<const char*>(
                                 weight + (size_t)nrow * EMB_D) + (size_t)tid * 16;
            __builtin_prefetch(pf, /*rw=*/0, /*locality=*/3);
        }

        // Per-row inverted-dropout scale (block-uniform scalar).
        const float u = row_mask_u[row];
        const float m = (u < keep) ? inv_keep : 0.0f;

        // Stream the 4 KB row: one global_load_b128 per lane.
        const v4f* __restrict__ wrow =
            reinterpret_cast<const v4f*>(weight + (size_t)row * EMB_D);
        v4f v = wrow[tid];
        v *= m;

        // Write-once output: non-temporal global_store_b128 (don't pollute L2).
        v4f* __restrict__ orow =
            reinterpret_cast<v4f*>(out + (size_t)t * EMB_D);
        __builtin_nontemporal_store(v, orow + tid);
    }
}

extern "C" void kernel_launch(void* const* d_in, const int* in_sizes, int n_in,
                              void* d_out, int out_size, void* d_ws, size_t ws_size,
                              hipStream_t stream)
{
    const int*   words      = reinterpret_cast<const int*>(d_in[0]);
    const float* weight     = reinterpret_cast<const float*>(d_in[1]);
    const float* row_mask_u = reinterpret_cast<const float*>(d_in[2]);
    float*       out        = reinterpret_cast<float*>(d_out);

    const int n_tokens = in_sizes[0];           // B*S = 16384

    // 4096 blocks -> ~4 tokens per block: enough loop trips for the prefetch
    // pipeline, and 4096 * 8 wave32s saturates the memory subsystem.
    int grid = n_tokens < 4096 ? (n_tokens > 0 ? n_tokens : 1) : 4096;

    embedded_dropout_gather<<<grid, 256, 0, stream>>>(
        words, weight, row_mask_u, out, n_tokens);
}